// DyadicSignature_67327907332419
// MI455X (gfx1250) — compile-verified
//
#include <hip/hip_runtime.h>

// ---------------------------------------------------------------------------
// Dyadic signature (level 3, 32 channels) + tree Chen combine + attention.
// Layout note: flattened signature = [a1 (32) | a2 (1024, i*32+j) | a3 (32768,
// i*1024+j*32+k)], which is exactly the reference's flatten order.
// a3 viewed as (r=(i,j), k) matrix (1024 x 32): every Chen step is a rank-1
// update, so a3 = U(1024x9) * V(9x32) -> WMMA f32 16x16x4 GEMM with K padded
// to 12. Tree combines use WMMA rank-1 (K=0 only) with C carrying the adds.
// ---------------------------------------------------------------------------

#define CH      32
#define NSTEP   9
#define KPAD    12
#define SIGC    33824
#define A2OFF   32
#define A3OFF   1056
#define NSEG    16
#define NBATCH  32
#define FRAMES  160
#define SEGLEN  10

typedef float v2f __attribute__((ext_vector_type(2)));
typedef float v8f __attribute__((ext_vector_type(8)));

// -------------------------- per-segment signature --------------------------
__global__ __launch_bounds__(256)
void sig_kernel(const float* __restrict__ x, float* __restrict__ sigs)
{
    __shared__ float V[KPAD * CH];    // (t, c)  increments, rows 9..11 zero
    __shared__ float W[KPAD * CH];    // (t, c)  w_t = a1pref + v/2
    __shared__ float A1P[KPAD * CH];  // (t, c)  exclusive prefix of v
    __shared__ float U[1024 * KPAD];  // (r, t)  rank-1 left factors

    const int tid = threadIdx.x;
    const int seg = blockIdx.x;          // 0..511
    const int b   = seg >> 4;
    const int g   = seg & 15;
    const float* xb = x + (size_t)b * (CH * FRAMES);
    const int f0 = g * SEGLEN;

    // increments v_t[c] = x[b,c,f0+t+1] - x[b,c,f0+t]
    for (int idx = tid; idx < KPAD * CH; idx += 256) {
        int t = idx >> 5, c = idx & 31;
        float v = 0.0f;
        if (t < NSTEP) v = xb[c * FRAMES + f0 + t + 1] - xb[c * FRAMES + f0 + t];
        V[idx] = v;
    }
    __syncthreads();

    // exclusive prefixes and w_t
    for (int idx = tid; idx < KPAD * CH; idx += 256) {
        int t = idx >> 5, c = idx & 31;
        float p = 0.0f;
        int lim = (t < NSTEP) ? t : NSTEP;
        for (int s = 0; s < lim; ++s) p += V[s * CH + c];
        A1P[idx] = p;
        W[idx] = (t < NSTEP) ? (p + 0.5f * V[idx]) : 0.0f;
    }
    __syncthreads();

    float* out = sigs + (size_t)seg * SIGC;

    // a1 (level 1) = total increment sum
    if (tid < CH)
        out[tid] = A1P[(NSTEP - 1) * CH + tid] + V[(NSTEP - 1) * CH + tid];

    // Build U (left factors for a3) and emit a2 directly.
    // u_t[(i,j)] = a2pref[i,j] + a1pref_t[i]*v_t[j]/2 + v_t[i]*v_t[j]/6
    const float inv6 = 1.0f / 6.0f;
    for (int r = tid; r < 1024; r += 256) {
        int i = r >> 5, j = r & 31;
        float a2run = 0.0f;
        #pragma unroll
        for (int t = 0; t < NSTEP; ++t) {
            float vi = V[t * CH + i], vj = V[t * CH + j];
            float u  = a2run + 0.5f * A1P[t * CH + i] * vj + inv6 * vi * vj;
            U[r * KPAD + t] = u;
            a2run += W[t * CH + i] * vj;
        }
        U[r * KPAD + 9]  = 0.0f;
        U[r * KPAD + 10] = 0.0f;
        U[r * KPAD + 11] = 0.0f;
        out[A2OFF + r] = a2run;          // level-2 signature
    }
    __syncthreads();

    // a3(1024x32) = U(1024x12) * V(12x32): 128 tiles of 16x16, K=12 -> 3 WMMA.
    const int wave = tid >> 5;
    const int lane = tid & 31;
    const int m    = lane & 15;
    const int hi   = lane >> 4;
    for (int q = 0; q < 16; ++q) {
        int tt = wave * 16 + q;
        int rt = tt >> 1, ct = tt & 1;
        int r0 = rt * 16, k0 = ct * 16;
        v8f acc = {};
        #pragma unroll
        for (int kk = 0; kk < KPAD; kk += 4) {
            v2f a, bb;
            // A 16x4: lane m = row; vgpr0 = K{0|2}, vgpr1 = K{1|3}
            a.x = U[(r0 + m) * KPAD + kk + hi * 2];
            a.y = U[(r0 + m) * KPAD + kk + hi * 2 + 1];
            // B 4x16: lane m = col; vgpr0 = row K{0|2}, vgpr1 = row K{1|3}
            bb.x = V[(kk + hi * 2) * CH + k0 + m];
            bb.y = V[(kk + hi * 2 + 1) * CH + k0 + m];
            acc = __builtin_amdgcn_wmma_f32_16x16x4_f32(
                false, a, false, bb, (short)0, acc, false, false);
        }
        #pragma unroll
        for (int d = 0; d < 8; ++d)
            out[A3OFF + (size_t)(r0 + d + hi * 8) * 32 + k0 + m] = acc[d];
    }
}

// ------------------------------ tree combine -------------------------------
// c = chen(a, b):
//   c1 = a1+b1
//   c2 = a2+b2 + a1 (x) b1                       (WMMA rank-1, C = a2+b2)
//   c3 = a3+b3 + a1 (x) b2 + a2 (x) b1
//     row-block i: c3[i] = a3[i]+b3[i] + a1[i]*b2 + outer(a2[i,:], b1)
//                                            (WMMA rank-1, C = the adds)
__global__ __launch_bounds__(256)
void combine_kernel(const float* __restrict__ src, float* __restrict__ dst,
                    int s_out, int dst_bstride)
{
    __shared__ float a1[CH], b1[CH];
    __shared__ float a2[1024], b2[1024];

    const int tid = threadIdx.x;
    const int wg  = blockIdx.x;
    const int b   = wg / s_out;
    const int p   = wg - b * s_out;
    const float* A = src + (size_t)(b * 2 * s_out + 2 * p) * SIGC;
    const float* B = A + SIGC;
    float* Dp = dst + (size_t)b * dst_bstride + (size_t)p * SIGC;

    for (int idx = tid; idx < 1024; idx += 256) {
        a2[idx] = A[A2OFF + idx];
        b2[idx] = B[A2OFF + idx];
    }
    if (tid < CH) {
        a1[tid] = A[tid];
        b1[tid] = B[tid];
        Dp[tid] = A[tid] + B[tid];       // c1
    }
    __syncthreads();

    const int wave = tid >> 5;
    const int lane = tid & 31;
    const int m    = lane & 15;
    const int hi   = lane >> 4;

    // c2: 4 tiles (waves 0..3; wave-uniform branch keeps EXEC all-ones)
    if (wave < 4) {
        int ti = wave >> 1, tj = wave & 1;
        v8f cfr;
        #pragma unroll
        for (int d = 0; d < 8; ++d) {
            int ii = ti * 16 + d + hi * 8, jj = tj * 16 + m;
            cfr[d] = a2[ii * 32 + jj] + b2[ii * 32 + jj];
        }
        v2f a, bb;
        a.x  = hi ? 0.0f : a1[ti * 16 + m];  a.y  = 0.0f;   // K=0 column only
        bb.x = hi ? 0.0f : b1[tj * 16 + m];  bb.y = 0.0f;   // K=0 row only
        v8f dfr = __builtin_amdgcn_wmma_f32_16x16x4_f32(
            false, a, false, bb, (short)0, cfr, false, false);
        #pragma unroll
        for (int d = 0; d < 8; ++d)
            Dp[A2OFF + (ti * 16 + d + hi * 8) * 32 + tj * 16 + m] = dfr[d];
    }

    // c3: 32 row-blocks x 4 tiles = 128 jobs over 8 waves
    for (int q = 0; q < 16; ++q) {
        int jj = wave * 16 + q;
        int i  = jj >> 2;
        int tq = jj & 3, ti = tq >> 1, tk = tq & 1;
        float a1i = a1[i];
        v8f cfr;
        #pragma unroll
        for (int d = 0; d < 8; ++d) {
            int j = ti * 16 + d + hi * 8;
            int k = tk * 16 + m;
            size_t off = (size_t)i * 1024 + j * 32 + k;
            cfr[d] = A[A3OFF + off] + B[A3OFF + off] + a1i * b2[j * 32 + k];
        }
        v2f a, bb;
        a.x  = hi ? 0.0f : a2[i * 32 + ti * 16 + m];  a.y  = 0.0f;
        bb.x = hi ? 0.0f : b1[tk * 16 + m];           bb.y = 0.0f;
        v8f dfr = __builtin_amdgcn_wmma_f32_16x16x4_f32(
            false, a, false, bb, (short)0, cfr, false, false);
        #pragma unroll
        for (int d = 0; d < 8; ++d)
            Dp[A3OFF + (size_t)i * 1024 + (ti * 16 + d + hi * 8) * 32
               + tk * 16 + m] = dfr[d];
    }
}

// ------------------------------- attention ---------------------------------
template <int NS>
__global__ __launch_bounds__(256)
void attn_kernel(const float* __restrict__ sigs,
                 const float* __restrict__ w, const float* __restrict__ bptr,
                 float* __restrict__ out)   // out pre-offset by column
{
    __shared__ float red[NS][256];
    __shared__ float scores[NS];

    const int tid = threadIdx.x;
    const int b   = blockIdx.x;
    const float* S = sigs + (size_t)b * NS * SIGC;

    float acc[NS];
    #pragma unroll
    for (int s = 0; s < NS; ++s) acc[s] = 0.0f;

    for (int c = tid; c < SIGC; c += 256) {
        float wc = w[c];
        #pragma unroll
        for (int s = 0; s < NS; ++s) acc[s] += S[(size_t)s * SIGC + c] * wc;
    }
    #pragma unroll
    for (int s = 0; s < NS; ++s) red[s][tid] = acc[s];
    __syncthreads();
    for (int off = 128; off > 0; off >>= 1) {
        if (tid < off) {
            #pragma unroll
            for (int s = 0; s < NS; ++s) red[s][tid] += red[s][tid + off];
        }
        __syncthreads();
    }
    if (tid == 0) {
        float bias = bptr[0];
        float mx = red[0][0] + bias;
        #pragma unroll
        for (int s = 1; s < NS; ++s) mx = fmaxf(mx, red[s][0] + bias);
        float sum = 0.0f;
        #pragma unroll
        for (int s = 0; s < NS; ++s) {
            float e = expf(red[s][0] + bias - mx);
            scores[s] = e;
            sum += e;
        }
        float inv = 1.0f / sum;
        #pragma unroll
        for (int s = 0; s < NS; ++s) scores[s] *= inv;
    }
    __syncthreads();

    for (int c = tid; c < SIGC; c += 256) {
        float o = 0.0f;
        #pragma unroll
        for (int s = 0; s < NS; ++s) o += scores[s] * S[(size_t)s * SIGC + c];
        out[(size_t)b * 5 * SIGC + c] = o;
    }
}

// ------------------------------- launcher ----------------------------------
extern "C" void kernel_launch(void* const* d_in, const int* in_sizes, int n_in,
                              void* d_out, int out_size, void* d_ws, size_t ws_size,
                              hipStream_t stream)
{
    (void)in_sizes; (void)n_in; (void)out_size; (void)ws_size;
    const float* x      = (const float*)d_in[0];   // (32, 32, 160)
    const float* attn_w = (const float*)d_in[1];   // (4, SIGC)
    const float* attn_b = (const float*)d_in[2];   // (4,)
    float* out  = (float*)d_out;                   // (32, 5, SIGC)
    float* bufA = (float*)d_ws;                    // 512 * SIGC  (level-4 sigs)
    float* bufB = bufA + (size_t)NBATCH * NSEG * SIGC;  // 256 * SIGC

    // level-4 signatures (16 segments per batch)
    sig_kernel<<<NBATCH * NSEG, 256, 0, stream>>>(x, bufA);
    attn_kernel<16><<<NBATCH, 256, 0, stream>>>(bufA, attn_w + 3 * SIGC,
                                                attn_b + 3, out + 4 * SIGC);
    combine_kernel<<<NBATCH * 8, 256, 0, stream>>>(bufA, bufB, 8, 8 * SIGC);
    attn_kernel<8><<<NBATCH, 256, 0, stream>>>(bufB, attn_w + 2 * SIGC,
                                               attn_b + 2, out + 3 * SIGC);
    combine_kernel<<<NBATCH * 4, 256, 0, stream>>>(bufB, bufA, 4, 4 * SIGC);
    attn_kernel<4><<<NBATCH, 256, 0, stream>>>(bufA, attn_w + 1 * SIGC,
                                               attn_b + 1, out + 2 * SIGC);
    combine_kernel<<<NBATCH * 2, 256, 0, stream>>>(bufA, bufB, 2, 2 * SIGC);
    attn_kernel<2><<<NBATCH, 256, 0, stream>>>(bufB, attn_w,
                                               attn_b, out + 1 * SIGC);
    // level-0 full signature straight into output column 0
    combine_kernel<<<NBATCH, 256, 0, stream>>>(bufB, out, 1, 5 * SIGC);
}